// SketchCritic_515396076091
// MI455X (gfx1250) — compile-verified
//
#include <hip/hip_runtime.h>
#include <math.h>

// MDN loss (Sketch-RNN): memory-bound streaming kernel + deterministic
// two-stage reduction. Final cross-block sum uses V_WMMA_F32_16X16X4_F32
// with an all-ones A matrix (layout-agnostic exact column-sum reduction).

typedef float v2f __attribute__((ext_vector_type(2)));
typedef float v8f __attribute__((ext_vector_type(8)));

#define SIGMA_MIN 0.01f
#define LOG_2PI   1.8378770664093453f
#define K   20
#define SEQ 2048
#define TPB 256

__global__ __launch_bounds__(TPB) void mdn_loss_kernel(
    const float* __restrict__ xs,      // (B,S,5)
    const float* __restrict__ logits,  // (B,S,K)
    const float* __restrict__ mus,     // (B,S,K,2)
    const float* __restrict__ sigx,    // (B,S,K)
    const float* __restrict__ sigy,    // (B,S,K)
    const float* __restrict__ sigxy,   // (B,S,K)
    const float* __restrict__ pen,     // (B,S,3)
    float* __restrict__ ws_pos,        // per-block sum of logp
    float* __restrict__ ws_pen,        // per-block sum of pen inner product
    int n_pairs)
{
    int p = blockIdx.x * TPB + threadIdx.x;
    float logp = 0.0f;
    float pen_inner = 0.0f;

    if (p < n_pairs) {
        size_t base5 = (size_t)p * 5;
        float x0  = xs[base5 + 0];
        float y0  = xs[base5 + 1];
        float pt0 = xs[base5 + 2];
        float pt1 = xs[base5 + 3];
        float pt2 = xs[base5 + 4];

        // roll(-1) along S within each batch row
        int s = p & (SEQ - 1);
        size_t pn = (s == SEQ - 1) ? (base5 - (size_t)(SEQ - 1) * 5) : (base5 + 5);
        float relx = xs[pn + 0] - x0;
        float rely = xs[pn + 1] - y0;

        const float4* lg4  = (const float4*)(logits + (size_t)p * K);
        const float4* sx4  = (const float4*)(sigx   + (size_t)p * K);
        const float4* sy4  = (const float4*)(sigy   + (size_t)p * K);
        const float4* sxy4 = (const float4*)(sigxy  + (size_t)p * K);
        const float4* mu4  = (const float4*)(mus    + (size_t)p * (K * 2));

        // Two running logsumexps: over logits, and over (logits + comp_logp).
        float m_l = -INFINITY, s_l = 0.0f;
        float m_a = -INFINITY, s_a = 0.0f;

        #pragma unroll
        for (int c = 0; c < K / 4; ++c) {
            float4 l4   = lg4[c];
            float4 sxv  = sx4[c];
            float4 syv  = sy4[c];
            float4 sxyv = sxy4[c];
            float4 mua  = mu4[2 * c];
            float4 mub  = mu4[2 * c + 1];

            float lv[4]   = {l4.x, l4.y, l4.z, l4.w};
            float sxa[4]  = {sxv.x, sxv.y, sxv.z, sxv.w};
            float sya[4]  = {syv.x, syv.y, syv.z, syv.w};
            float sxya[4] = {sxyv.x, sxyv.y, sxyv.z, sxyv.w};
            float mx[4]   = {mua.x, mua.z, mub.x, mub.z};
            float my[4]   = {mua.y, mua.w, mub.y, mub.w};

            #pragma unroll
            for (int j = 0; j < 4; ++j) {
                float sx  = fmaxf(sxa[j], SIGMA_MIN);
                float sy  = fmaxf(sya[j], SIGMA_MIN);
                float sxy = fminf(fmaxf(sxya[j], -SIGMA_MIN), SIGMA_MIN);
                // fast reciprocals (v_rcp_f32) instead of IEEE divide sequences
                float z1  = (relx - mx[j]) * __builtin_amdgcn_rcpf(sx);
                float z2  = ((rely - my[j]) - sxy * z1) * __builtin_amdgcn_rcpf(sy);
                // log(sx)+log(sy) == log(sx*sy): one v_log_f32 instead of two
                float clp = -0.5f * (z1 * z1 + z2 * z2) - LOG_2PI
                            - __logf(sx * sy);
                float l = lv[j];
                float a = l + clp;

                float m2 = fmaxf(m_l, l);
                s_l = s_l * __expf(m_l - m2) + __expf(l - m2);
                m_l = m2;

                float m3 = fmaxf(m_a, a);
                s_a = s_a * __expf(m_a - m3) + __expf(a - m3);
                m_a = m3;
            }
        }
        // logp = LSE(logits + comp) - LSE(logits)
        logp = (m_a + __logf(s_a)) - (m_l + __logf(s_l));

        // pen term: sum(pen_true * log_softmax(pen_pred))
        size_t base3 = (size_t)p * 3;
        float q0 = pen[base3 + 0], q1 = pen[base3 + 1], q2 = pen[base3 + 2];
        float pm  = fmaxf(q0, fmaxf(q1, q2));
        float lse = pm + __logf(__expf(q0 - pm) + __expf(q1 - pm) + __expf(q2 - pm));
        pen_inner = pt0 * (q0 - lse) + pt1 * (q1 - lse) + pt2 * (q2 - lse);
    }

    // wave32 reduction
    #pragma unroll
    for (int off = 16; off > 0; off >>= 1) {
        logp      += __shfl_xor(logp, off, 32);
        pen_inner += __shfl_xor(pen_inner, off, 32);
    }
    __shared__ float sp[TPB / 32];
    __shared__ float sq[TPB / 32];
    int lane = threadIdx.x & 31;
    int w    = threadIdx.x >> 5;
    if (lane == 0) { sp[w] = logp; sq[w] = pen_inner; }
    __syncthreads();
    if (threadIdx.x == 0) {
        float tp = 0.0f, tq = 0.0f;
        #pragma unroll
        for (int i = 0; i < TPB / 32; ++i) { tp += sp[i]; tq += sq[i]; }
        ws_pos[blockIdx.x] = tp;
        ws_pen[blockIdx.x] = tq;
    }
}

// Final reduction over per-block partials using V_WMMA_F32_16X16X4_F32.
// A = all ones (16x4) => D[m,n] = sum_k B[k,n] + C[m,n]: each WMMA reduces
// 64 fp32 values into 16 accumulating column sums; row M=0 of D lives in
// c[0] on lanes 0..15. Summing those 16 lanes yields the exact total,
// independent of the B operand's lane->(k,n) mapping. Single wave, EXEC all 1s.
// Hot loop is branch-free: B operands loaded as contiguous v2f (b64 loads);
// any non-multiple-of-64 tail is folded in serially by lane 0 (empty for the
// real problem size, nblocks = 2048).
__global__ __launch_bounds__(32) void finalize_wmma(
    const float* __restrict__ ws_pos,
    const float* __restrict__ ws_pen,
    float* __restrict__ out,
    int nblocks, float inv_count)
{
    int lane = threadIdx.x;  // 0..31
    v2f a; a[0] = 1.0f; a[1] = 1.0f;
    v8f cp = {};
    v8f cq = {};

    const v2f* pos2 = (const v2f*)ws_pos;
    const v2f* pen2 = (const v2f*)ws_pen;

    int full_rounds = nblocks >> 6;  // 64 values per round
    for (int r = 0; r < full_rounds; ++r) {
        v2f bp = pos2[(r << 5) + lane];
        v2f bq = pen2[(r << 5) + lane];
        cp = __builtin_amdgcn_wmma_f32_16x16x4_f32(
            false, a, false, bp, (short)0, cp, false, false);
        cq = __builtin_amdgcn_wmma_f32_16x16x4_f32(
            false, a, false, bq, (short)0, cq, false, false);
    }

    __shared__ float rp[32];
    __shared__ float rq[32];
    rp[lane] = cp[0];
    rq[lane] = cq[0];
    __syncthreads();
    if (lane == 0) {
        float tp = 0.0f, tq = 0.0f;
        #pragma unroll
        for (int i = 0; i < 16; ++i) { tp += rp[i]; tq += rq[i]; }
        // serial tail for nblocks not divisible by 64 (empty in practice)
        for (int i = full_rounds << 6; i < nblocks; ++i) {
            tp += ws_pos[i];
            tq += ws_pen[i];
        }
        out[0] = -tp * inv_count;  // position_loss = -mean(logp)
        out[1] = -tq * inv_count;  // pen_loss
    }
}

extern "C" void kernel_launch(void* const* d_in, const int* in_sizes, int n_in,
                              void* d_out, int out_size, void* d_ws, size_t ws_size,
                              hipStream_t stream)
{
    const float* xs     = (const float*)d_in[0];
    const float* logits = (const float*)d_in[1];
    const float* mus    = (const float*)d_in[2];
    const float* sigx   = (const float*)d_in[3];
    const float* sigy   = (const float*)d_in[4];
    const float* sigxy  = (const float*)d_in[5];
    const float* pen    = (const float*)d_in[6];
    float* out = (float*)d_out;

    int n_pairs = in_sizes[0] / 5;               // B*S = 524288
    int nblocks = (n_pairs + TPB - 1) / TPB;     // 2048 (multiple of 64)

    float* ws_pos = (float*)d_ws;
    float* ws_pen = ws_pos + nblocks;

    mdn_loss_kernel<<<nblocks, TPB, 0, stream>>>(
        xs, logits, mus, sigx, sigy, sigxy, pen, ws_pos, ws_pen, n_pairs);
    finalize_wmma<<<1, 32, 0, stream>>>(
        ws_pos, ws_pen, out, nblocks, 1.0f / (float)n_pairs);
}